// TrajLoss_6141803233879
// MI455X (gfx1250) — compile-verified
//
#include <hip/hip_runtime.h>

// ---------------------------------------------------------------------------
// TrajLoss: sum_i (x[i]-pred[i])^2 + (y[i]-pred[i+1])^2, N = 16.7M, f32.
// Memory-bound (~0.5 flop/byte): stream 192MB at HBM rate, reduce with
// V_WMMA_F32_16X16X4_F32 (full-precision f32 WMMA) at wave level.
// ---------------------------------------------------------------------------

typedef __attribute__((ext_vector_type(2))) float v2f;
typedef __attribute__((ext_vector_type(8))) float v8f;

// Exact 32-lane f32 sum via WMMA.
// A layout (16x4 f32): lanes 0-15 -> (M=lane, K=0) in VGPR0, (K=1) in VGPR1;
//                      lanes 16-31 -> (M=lane-16, K=2) in VGPR0, (K=3) in VGPR1.
// Put p in VGPR0, 0 in VGPR1; B = all ones (B layout irrelevant for constant 1).
// => D[m][n] = p[m] + p[m+16] for all n.
// D layout: VGPR r: lanes 0-15 hold M=r, lanes 16-31 hold M=r+8.
// Sum of 8 D-VGPRs: lanes<16 = sum over rows 0-7, lanes>=16 = rows 8-15.
// One xor-16 exchange gives the full 32-lane total in every lane.
__device__ __forceinline__ float wave_reduce_wmma(float p) {
    v2f a; a[0] = p;    a[1] = 0.0f;
    v2f b; b[0] = 1.0f; b[1] = 1.0f;
    v8f c = {};
    c = __builtin_amdgcn_wmma_f32_16x16x4_f32(
        /*neg_a=*/false, a, /*neg_b=*/false, b,
        /*c_mod=*/(short)0, c, /*reuse_a=*/false, /*reuse_b=*/false);
    float s = ((c[0] + c[1]) + (c[2] + c[3])) + ((c[4] + c[5]) + (c[6] + c[7]));
    s += __shfl_xor(s, 16, 32);
    return s;
}

// Deterministic block reduction: wave WMMA-reduce -> LDS -> sequential combine.
// Result valid on threadIdx.x == 0 only.
__device__ __forceinline__ float block_reduce(float p, float* lds) {
    float w = wave_reduce_wmma(p);
    const int lane = threadIdx.x & 31;
    const int wid  = threadIdx.x >> 5;
    if (lane == 0) lds[wid] = w;
    __syncthreads();
    float total = 0.0f;
    if (threadIdx.x == 0) {
        const int nw = blockDim.x >> 5;
        for (int i = 0; i < nw; ++i) total += lds[i];
    }
    return total;
}

__global__ void trajloss_partial_kernel(const float* __restrict__ pred,
                                        const float* __restrict__ tx,
                                        const float* __restrict__ ty,
                                        float* __restrict__ partial,
                                        int n) {
    __shared__ float lds[32];
    const int n4     = n >> 2;                      // whole float4 chunks
    const int gid    = blockIdx.x * blockDim.x + threadIdx.x;
    const int stride = gridDim.x * blockDim.x;

    const float4* __restrict__ x4 = reinterpret_cast<const float4*>(tx);
    const float4* __restrict__ y4 = reinterpret_cast<const float4*>(ty);
    const float4* __restrict__ p4 = reinterpret_cast<const float4*>(pred);

    float acc = 0.0f;
    for (int v = gid; v < n4; v += stride) {
        // Prefetch next grid-stride iteration (speculative, OOB silently dropped).
        __builtin_prefetch(&x4[v + stride], 0, 0);
        __builtin_prefetch(&y4[v + stride], 0, 0);
        __builtin_prefetch(&p4[v + stride], 0, 0);

        const float4 x = x4[v];
        const float4 y = y4[v];
        const float4 p = p4[v];                     // pred[4v .. 4v+3], 16B aligned
        const float  pn = pred[4 * v + 4];          // pred[4v+4] (pred has n+1 elems)

        const float dx0 = x.x - p.x, dx1 = x.y - p.y, dx2 = x.z - p.z, dx3 = x.w - p.w;
        const float dy0 = y.x - p.y, dy1 = y.y - p.z, dy2 = y.z - p.w, dy3 = y.w - pn;

        acc = fmaf(dx0, dx0, acc); acc = fmaf(dy0, dy0, acc);
        acc = fmaf(dx1, dx1, acc); acc = fmaf(dy1, dy1, acc);
        acc = fmaf(dx2, dx2, acc); acc = fmaf(dy2, dy2, acc);
        acc = fmaf(dx3, dx3, acc); acc = fmaf(dy3, dy3, acc);
    }
    // Scalar tail (n not multiple of 4).
    for (int i = (n4 << 2) + gid; i < n; i += stride) {
        const float dx = tx[i] - pred[i];
        const float dy = ty[i] - pred[i + 1];
        acc = fmaf(dx, dx, acc);
        acc = fmaf(dy, dy, acc);
    }

    const float t = block_reduce(acc, lds);
    if (threadIdx.x == 0) partial[blockIdx.x] = t;
}

__global__ void trajloss_final_kernel(const float* __restrict__ partial,
                                      int nb,
                                      float* __restrict__ out) {
    __shared__ float lds[32];
    float acc = 0.0f;
    for (int i = threadIdx.x; i < nb; i += blockDim.x) acc += partial[i];
    const float t = block_reduce(acc, lds);
    if (threadIdx.x == 0) out[0] = t;
}

extern "C" void kernel_launch(void* const* d_in, const int* in_sizes, int n_in,
                              void* d_out, int out_size, void* d_ws, size_t ws_size,
                              hipStream_t stream) {
    const float* pred = (const float*)d_in[0];   // N+1 floats
    const float* tx   = (const float*)d_in[1];   // N floats
    const float* ty   = (const float*)d_in[2];   // N floats
    const int    n    = in_sizes[1];

    float* partial = (float*)d_ws;
    float* out     = (float*)d_out;

    // 2048 blocks x 256 threads (8 waves/block) = 512K threads; 16.7M elems
    // -> ~8 float4 iterations/thread: enough in-flight loads to saturate HBM.
    int blocks = 2048;
    const int maxBlocksByWs = (int)(ws_size / sizeof(float));
    if (blocks > maxBlocksByWs) blocks = maxBlocksByWs;
    if (blocks < 1) blocks = 1;

    trajloss_partial_kernel<<<blocks, 256, 0, stream>>>(pred, tx, ty, partial, n);
    trajloss_final_kernel<<<1, 256, 0, stream>>>(partial, blocks, out);
}